// GWPooling2D_89687507076120
// MI455X (gfx1250) — compile-verified
//
#include <hip/hip_runtime.h>
#include <stdint.h>

// ---------------------------------------------------------------------------
// GWPooling2D for MI455X (gfx1250).  fp32 WMMA (V_WMMA_F32_16X16X4_F32) for
// all GEMMs; double-buffered async global->LDS (GLOBAL_LOAD_ASYNC_TO_LDS_B128)
// staging in both the expm complex GEMM (the FLOP-dominant stage, ~0.5 TFLOP)
// and the final 8192x4096x400 GEMM.
// ---------------------------------------------------------------------------

typedef __attribute__((ext_vector_type(2))) float v2f;
typedef __attribute__((ext_vector_type(8))) float v8f;

#define TWO_PI_F 6.28318530717958647692f

__device__ __forceinline__ v8f wmma_f32(v2f a, v2f b, v8f c) {
  // (neg_a, A, neg_b, B, c_mod, C, reuse_a, reuse_b)
  return __builtin_amdgcn_wmma_f32_16x16x4_f32(false, a, false, b, (short)0, c,
                                               false, false);
}

// ======================= Stage A: signal -> spectrum =======================
// 32 blocks = (c,s) planes, 576 threads = one 24x24 plane.

__global__ void k_fft_filter(const float* __restrict__ sig,
                             float* __restrict__ gr, float* __restrict__ gi) {
  __shared__ float ss[576];
  __shared__ float c24[24], s24[24];
  const int b = blockIdx.x;
  const int tid = threadIdx.x;
  ss[tid] = sig[b * 576 + tid];
  if (tid < 24) {
    float th = TWO_PI_F * (float)tid / 24.0f;
    c24[tid] = cosf(th);
    s24[tid] = sinf(th);
  }
  __syncthreads();
  const int u = tid / 24, v = tid % 24;
  float ar = 0.f, ai = 0.f;
  for (int x = 0; x < 24; ++x) {
    int pu = (u * x) % 24;
    for (int y = 0; y < 24; ++y) {
      int p = (pu + v * y) % 24;
      float s1 = ss[x * 24 + y];
      ar += s1 * c24[p];   // real input * exp(-i theta)
      ai -= s1 * s24[p];
    }
  }
  // H[u,v] = 1 / (5 * r_ifftshift): r[p,q] = 1+(p-11)^2+(q-11)^2, p=(u+12)%24
  int ru = ((u + 12) % 24) - 11;
  int rv = ((v + 12) % 24) - 11;
  float h = 1.0f / (5.0f * (1.0f + (float)(ru * ru + rv * rv)));
  gr[b * 576 + tid] = ar * h;
  gi[b * 576 + tid] = ai * h;
}

__global__ void k_ifft_window(const float* __restrict__ gr,
                              const float* __restrict__ gi,
                              float* __restrict__ wr, float* __restrict__ wi) {
  __shared__ float sr[576], si[576];
  __shared__ float c24[24], s24[24];
  const int b = blockIdx.x;
  const int tid = threadIdx.x;
  sr[tid] = gr[b * 576 + tid];
  si[tid] = gi[b * 576 + tid];
  if (tid < 24) {
    float th = TWO_PI_F * (float)tid / 24.0f;
    c24[tid] = cosf(th);
    s24[tid] = sinf(th);
  }
  __syncthreads();
  const int x = tid / 24, y = tid % 24;
  float ar = 0.f, ai = 0.f;
  for (int u = 0; u < 24; ++u) {
    int pu = (x * u) % 24;
    for (int v = 0; v < 24; ++v) {
      int p = (pu + y * v) % 24;
      float c = c24[p], s = s24[p];  // exp(+i theta)
      ar += sr[u * 24 + v] * c - si[u * 24 + v] * s;
      ai += sr[u * 24 + v] * s + si[u * 24 + v] * c;
    }
  }
  float winx = 0.5f * (1.0f - cosf(TWO_PI_F * (float)x / 24.0f));
  float winy = 0.5f * (1.0f - cosf(TWO_PI_F * (float)y / 24.0f));
  float w = winx * winy * (1.0f / 576.0f);
  wr[b * 576 + tid] = ar * w;
  wi[b * 576 + tid] = ai * w;
}

__global__ void k_fft_shift(const float* __restrict__ wr,
                            const float* __restrict__ wi,
                            float* __restrict__ outr,
                            float* __restrict__ outi) {
  __shared__ float sr[576], si[576];
  __shared__ float c24[24], s24[24];
  const int b = blockIdx.x;
  const int tid = threadIdx.x;
  sr[tid] = wr[b * 576 + tid];
  si[tid] = wi[b * 576 + tid];
  if (tid < 24) {
    float th = TWO_PI_F * (float)tid / 24.0f;
    c24[tid] = cosf(th);
    s24[tid] = sinf(th);
  }
  __syncthreads();
  const int u = tid / 24, v = tid % 24;
  float ar = 0.f, ai = 0.f;
  for (int x = 0; x < 24; ++x) {
    int pu = (u * x) % 24;
    for (int y = 0; y < 24; ++y) {
      int p = (pu + v * y) % 24;
      float c = c24[p], s = s24[p];  // exp(-i theta)
      ar += sr[x * 24 + y] * c + si[x * 24 + y] * s;
      ai += si[x * 24 + y] * c - sr[x * 24 + y] * s;
    }
  }
  // spectrum = roll(F,(12,12)): write F[u,v] at ((u+12)%24,(v+12)%24)
  int uu = (u + 12) % 24, vv = (v + 12) % 24;
  outr[b * 576 + uu * 24 + vv] = ar;
  outi[b * 576 + uu * 24 + vv] = ai;
}

// ======================= Stage B: build scaled A matrix ====================
#define NMAT 576
#define PLANE 331776          // 576*576
#define NCH 16
#define EXPM_S 10             // fixed scaling: A/2^10
#define EXPM_SCALE (1.0f / 1024.0f)

__global__ void k_build_B(const float* __restrict__ sre,
                          const float* __restrict__ sim,
                          float* __restrict__ br, float* __restrict__ bi) {
  int idx = blockIdx.x * blockDim.x + threadIdx.x;  // 16*576*576 exactly
  int c = idx / PLANE;
  int rem = idx % PLANE;
  int m = rem / NMAT;
  int n = rem % NMAT;
  int A = m / 24, Bq = m % 24;
  int i = n / 24, j = n % 24;
  int u0 = 23 - A + i;
  int v0 = 23 - Bq + j;
  float s0r = 0.f, s0i = 0.f, s1r = 0.f, s1i = 0.f;
  if (u0 >= 12 && u0 < 36 && v0 >= 12 && v0 < 36) {
    int si_ = (u0 - 12) * 24 + (v0 - 12);
    int base = c * 1152;
    s0r = sre[base + si_];
    s0i = sim[base + si_];
    s1r = sre[base + 576 + si_];
    s1i = sim[base + 576 + si_];
  }
  float kx = (float)(i - 11);
  float ky = (float)(j - 11);
  // mat = -1j*(S0*kx + S1*ky), then scale by 2^-s
  br[idx] = (s0i * kx + s1i * ky) * EXPM_SCALE;
  bi[idx] = -(s0r * kx + s1r * ky) * EXPM_SCALE;
}

__global__ void k_init_T(const float* __restrict__ br,
                         const float* __restrict__ bi,
                         float* __restrict__ tr, float* __restrict__ ti) {
  int idx = blockIdx.x * blockDim.x + threadIdx.x;
  int rem = idx % PLANE;
  int m = rem / NMAT, n = rem % NMAT;
  tr[idx] = br[idx] * (1.0f / 12.0f) + ((m == n) ? 1.0f : 0.0f);
  ti[idx] = bi[idx] * (1.0f / 12.0f);
}

// ============ Complex 576x576 GEMM via fp32 WMMA (per channel) =============
// C = alpha*(A@B) [+ I].  Block = 256 threads = 8 waves; 64x64 block tile,
// K-panel 16 staged in LDS (re+im planes) with double-buffered async copies.
// Waves: 2 (rows of 32) x 4 (cols of 16); each wave: 2 x (P,Q,R,S) v8f accums.
// Grid = 16 channels * 81 blocks (9x9 tiles of 64).

__global__ __launch_bounds__(256) void cgemm_wmma(
    const float* __restrict__ Ar, const float* __restrict__ Ai,
    const float* __restrict__ Br, const float* __restrict__ Bi,
    float* __restrict__ Cr, float* __restrict__ Ci, float alpha, int addI) {
  // A: [buf][plane][row 64][20 floats, 16 used] ; B: [buf][plane][k 16][64]
  __shared__ float sA[2 * 2 * 64 * 20];  // 20 KB
  __shared__ float sB[2 * 2 * 16 * 64];  // 16 KB

  const int t = threadIdx.x;
  const int ch = blockIdx.x / 81;
  const int blk = blockIdx.x % 81;
  const int bm = (blk / 9) * 64;
  const int bn = (blk % 9) * 64;

  const float* ar = Ar + (size_t)ch * PLANE;
  const float* ai = Ai + (size_t)ch * PLANE;
  const float* br = Br + (size_t)ch * PLANE;
  const float* bi = Bi + (size_t)ch * PLANE;
  float* cr = Cr + (size_t)ch * PLANE;
  float* ci = Ci + (size_t)ch * PLANE;

  const unsigned ldsA = (unsigned)(size_t)&sA[0];
  const unsigned ldsB = (unsigned)(size_t)&sB[0];

  // A panel copy: thread t -> row t>>2 (0..63), kquad t&3 (one B128 per plane)
  const unsigned aOffBase =
      (unsigned)(((bm + (t >> 2)) * NMAT + (t & 3) * 4) * 4);
  const unsigned ldsAslot = (unsigned)((t >> 2) * 80 + (t & 3) * 16);
  // B panel copy: thread t -> krow t>>4 (0..15), colquad t&15
  const unsigned bOffBase =
      (unsigned)(((t >> 4) * NMAT + bn + (t & 15) * 4) * 4);
  const unsigned ldsBslot = (unsigned)((t >> 4) * 256 + (t & 15) * 16);

  auto issue = [&](int k0, int buf) {
    unsigned va = aOffBase + (unsigned)k0 * 4u;      // +k0 floats
    unsigned vb = bOffBase + (unsigned)k0 * 2304u;   // +k0 rows of 576 floats
    unsigned laR = ldsA + (unsigned)buf * 10240u + ldsAslot;
    unsigned laI = laR + 5120u;
    unsigned lbR = ldsB + (unsigned)buf * 8192u + ldsBslot;
    unsigned lbI = lbR + 4096u;
    asm volatile("global_load_async_to_lds_b128 %0, %1, %2"
                 :: "v"(laR), "v"(va), "s"(ar) : "memory");
    asm volatile("global_load_async_to_lds_b128 %0, %1, %2"
                 :: "v"(laI), "v"(va), "s"(ai) : "memory");
    asm volatile("global_load_async_to_lds_b128 %0, %1, %2"
                 :: "v"(lbR), "v"(vb), "s"(br) : "memory");
    asm volatile("global_load_async_to_lds_b128 %0, %1, %2"
                 :: "v"(lbI), "v"(vb), "s"(bi) : "memory");
  };

  const int lane = t & 31;
  const int wave = t >> 5;
  const int wm = wave & 1;    // 0..1 -> 32-row slabs
  const int wn = wave >> 1;   // 0..3 -> 16-col slabs
  const int mb = wm * 32;
  const int nbl = wn * 16;
  const int ml = lane & 15;
  const int khalf = 2 * (lane >> 4);

  v8f accP[2] = {{}, {}}, accQ[2] = {{}, {}};
  v8f accR[2] = {{}, {}}, accS[2] = {{}, {}};

  issue(0, 0);
  asm volatile("s_wait_asynccnt 0" ::: "memory");
  __syncthreads();

  for (int ks = 0; ks < 36; ++ks) {
    const int buf = ks & 1;
    if (ks < 35) issue((ks + 1) * 16, buf ^ 1);

    const int aB = buf * 2560;  // floats
    const int bB = buf * 2048;
#pragma unroll
    for (int kk = 0; kk < 4; ++kk) {
      int kb = kk * 4 + khalf;
      v2f ar0, ar1, ai0, ai1, brf, bif;
      ar0.x = sA[aB + (mb + ml) * 20 + kb];
      ar0.y = sA[aB + (mb + ml) * 20 + kb + 1];
      ar1.x = sA[aB + (mb + 16 + ml) * 20 + kb];
      ar1.y = sA[aB + (mb + 16 + ml) * 20 + kb + 1];
      ai0.x = sA[aB + 1280 + (mb + ml) * 20 + kb];
      ai0.y = sA[aB + 1280 + (mb + ml) * 20 + kb + 1];
      ai1.x = sA[aB + 1280 + (mb + 16 + ml) * 20 + kb];
      ai1.y = sA[aB + 1280 + (mb + 16 + ml) * 20 + kb + 1];
      brf.x = sB[bB + kb * 64 + nbl + ml];
      brf.y = sB[bB + (kb + 1) * 64 + nbl + ml];
      bif.x = sB[bB + 1024 + kb * 64 + nbl + ml];
      bif.y = sB[bB + 1024 + (kb + 1) * 64 + nbl + ml];
      accP[0] = wmma_f32(ar0, brf, accP[0]);
      accP[1] = wmma_f32(ar1, brf, accP[1]);
      accQ[0] = wmma_f32(ai0, bif, accQ[0]);
      accQ[1] = wmma_f32(ai1, bif, accQ[1]);
      accR[0] = wmma_f32(ar0, bif, accR[0]);
      accR[1] = wmma_f32(ar1, bif, accR[1]);
      accS[0] = wmma_f32(ai0, brf, accS[0]);
      accS[1] = wmma_f32(ai1, brf, accS[1]);
    }
    asm volatile("s_wait_asynccnt 0" ::: "memory");
    __syncthreads();
  }

  const int rhalf = 8 * (lane >> 4);
#pragma unroll
  for (int f = 0; f < 2; ++f) {
#pragma unroll
    for (int v = 0; v < 8; ++v) {
      int row = bm + mb + f * 16 + v + rhalf;
      int col = bn + nbl + ml;
      float re = alpha * (accP[f][v] - accQ[f][v]);
      if (addI && row == col) re += 1.0f;
      float im = alpha * (accR[f][v] + accS[f][v]);
      cr[row * NMAT + col] = re;
      ci[row * NMAT + col] = im;
    }
  }
}

// ======================= Stage D: transform_to_map =========================
#define MAPN 1638400  // 16*16*16*20*20

__global__ void k_slice_roll(const float* __restrict__ tr,
                             const float* __restrict__ ti,
                             float* __restrict__ pr, float* __restrict__ pi) {
  int idx = blockIdx.x * blockDim.x + threadIdx.x;
  if (idx >= MAPN) return;
  int c = idx / 102400;
  int r = idx % 102400;
  int o0 = r / 6400;
  int r2 = r % 6400;
  int o1 = r2 / 400;
  int r3 = r2 % 400;
  int i0 = r3 / 20;
  int i1 = r3 % 20;
  int so0 = (o0 + 7) & 15;     // (o0-9) mod 16
  int so1 = (o1 + 7) & 15;
  int si0 = (i0 + 9) % 20;     // (i0-11) mod 20
  int si1 = (i1 + 9) % 20;
  int m = (so0 + 5) * 24 + (so1 + 5);
  int n = (si0 + 3) * 24 + (si1 + 3);
  size_t src = (size_t)c * PLANE + m * NMAT + n;
  pr[idx] = tr[src];
  pi[idx] = ti[src];
}

// Generic 1-D DFT pass on [outer, N, inner]; sign=+1 -> ifft direction.
__global__ void k_dft(const float* __restrict__ ir, const float* __restrict__ ii,
                      float* __restrict__ outr, float* __restrict__ outi,
                      int N, int inner, float sign, float scale) {
  int idx = blockIdx.x * blockDim.x + threadIdx.x;
  if (idx >= MAPN) return;
  int ni = N * inner;
  int o = idx / ni;
  int r = idx % ni;
  int a = r / inner;
  int in_i = r % inner;
  int base = o * ni + in_i;
  float th = sign * TWO_PI_F * (float)a / (float)N;
  float rc = cosf(th), rs = sinf(th);
  float cr = 1.0f, ci_ = 0.0f;
  float ar = 0.f, ai_ = 0.f;
  for (int b = 0; b < N; ++b) {
    float xr = ir[base + b * inner];
    float xi = ii[base + b * inner];
    ar += xr * cr - xi * ci_;
    ai_ += xr * ci_ + xi * cr;
    float nc = cr * rc - ci_ * rs;
    ci_ = cr * rs + ci_ * rc;
    cr = nc;
  }
  outr[idx] = ar * scale;
  outi[idx] = ai_ * scale;
}

__global__ void k_pack_W(const float* __restrict__ pr, float* __restrict__ W) {
  int idx = blockIdx.x * blockDim.x + threadIdx.x;
  if (idx >= MAPN) return;
  int c = idx / 102400;
  int r = idx % 102400;
  int o0 = r / 6400;
  int r2 = r % 6400;
  int o1 = r2 / 400;
  int r3 = r2 % 400;
  int i0 = r3 / 20;
  int i1 = r3 % 20;
  // W[k=400][n=4096] row-major, n = c*256 + o0*16 + o1
  W[(i0 * 20 + i1) * 4096 + c * 256 + o0 * 16 + o1] = pr[idx];
}

// ===================== Final GEMM: Y = X(8192x400) @ W(400x4096) ===========
// 128x64 block tile, 8 waves of 32x32 (2x2 WMMA frags), K staged in LDS via
// double-buffered async global->LDS copies.  A tile padded to 20 floats/row.

__global__ __launch_bounds__(256) void sgemm_wmma_async(
    const float* __restrict__ X, const float* __restrict__ W,
    float* __restrict__ Y) {
  __shared__ float sA[2][128 * 20];  // 10KB per stage (16 of 20 used per row)
  __shared__ float sB[2][16 * 64];   // 4KB per stage

  const int t = threadIdx.x;
  const int bm = blockIdx.y * 128;
  const int bn = blockIdx.x * 64;

  const unsigned ldsA = (unsigned)(size_t)&sA[0][0];
  const unsigned ldsB = (unsigned)(size_t)&sB[0][0];

  // A-tile copy: groups of 4 floats; group g -> row g>>2, kquad g&3
  const int g0 = t, g1 = t + 256;
  const unsigned aBase0 = (unsigned)(((bm + (g0 >> 2)) * 400 + (g0 & 3) * 4) * 4);
  const unsigned aBase1 = (unsigned)(((bm + (g1 >> 2)) * 400 + (g1 & 3) * 4) * 4);
  const unsigned aSlot0 = (unsigned)((g0 >> 2) * 80 + (g0 & 3) * 16);
  const unsigned aSlot1 = (unsigned)((g1 >> 2) * 80 + (g1 & 3) * 16);
  // B-tile copy: group t -> krow t>>4, colquad t&15
  const unsigned bBase = (unsigned)((((t >> 4) * 4096) + bn + (t & 15) * 4) * 4);

  auto issue = [&](int k0, int buf) {
    unsigned la0 = ldsA + (unsigned)buf * 10240u + aSlot0;
    unsigned la1 = ldsA + (unsigned)buf * 10240u + aSlot1;
    unsigned lb = ldsB + (unsigned)buf * 4096u + (unsigned)t * 16u;
    unsigned va0 = aBase0 + (unsigned)k0 * 4u;          // +k0 floats
    unsigned va1 = aBase1 + (unsigned)k0 * 4u;
    unsigned vb = bBase + (unsigned)k0 * 16384u;        // +k0 rows of 4096
    asm volatile("global_load_async_to_lds_b128 %0, %1, %2"
                 :: "v"(la0), "v"(va0), "s"(X) : "memory");
    asm volatile("global_load_async_to_lds_b128 %0, %1, %2"
                 :: "v"(la1), "v"(va1), "s"(X) : "memory");
    asm volatile("global_load_async_to_lds_b128 %0, %1, %2"
                 :: "v"(lb), "v"(vb), "s"(W) : "memory");
  };

  const int lane = t & 31;
  const int wave = t >> 5;
  const int wm = wave >> 1;     // 0..3 -> rows
  const int wn = wave & 1;      // 0..1 -> cols
  const int mb = wm * 32;
  const int nb = wn * 32;
  const int ml = lane & 15;
  const int khalf = 2 * (lane >> 4);

  v8f acc00 = {}, acc01 = {}, acc10 = {}, acc11 = {};

  // prologue: stage 0
  issue(0, 0);
  asm volatile("s_wait_asynccnt 0" ::: "memory");
  __syncthreads();

  for (int ks = 0; ks < 25; ++ks) {
    const int buf = ks & 1;
    if (ks < 24) issue((ks + 1) * 16, buf ^ 1);

#pragma unroll
    for (int kk = 0; kk < 4; ++kk) {
      int kbase = kk * 4 + khalf;
      v2f a0, a1, b0, b1;
      a0.x = sA[buf][(mb + ml) * 20 + kbase];
      a0.y = sA[buf][(mb + ml) * 20 + kbase + 1];
      a1.x = sA[buf][(mb + 16 + ml) * 20 + kbase];
      a1.y = sA[buf][(mb + 16 + ml) * 20 + kbase + 1];
      b0.x = sB[buf][kbase * 64 + nb + ml];
      b0.y = sB[buf][(kbase + 1) * 64 + nb + ml];
      b1.x = sB[buf][kbase * 64 + nb + 16 + ml];
      b1.y = sB[buf][(kbase + 1) * 64 + nb + 16 + ml];
      acc00 = wmma_f32(a0, b0, acc00);
      acc01 = wmma_f32(a0, b1, acc01);
      acc10 = wmma_f32(a1, b0, acc10);
      acc11 = wmma_f32(a1, b1, acc11);
    }
    asm volatile("s_wait_asynccnt 0" ::: "memory");
    __syncthreads();
  }

  const int rhalf = 8 * (lane >> 4);
#pragma unroll
  for (int v = 0; v < 8; ++v) {
    int r0 = bm + mb + v + rhalf;
    int c0 = bn + nb + ml;
    Y[(size_t)r0 * 4096 + c0] = acc00[v];
    Y[(size_t)r0 * 4096 + c0 + 16] = acc01[v];
    Y[(size_t)(r0 + 16) * 4096 + c0] = acc10[v];
    Y[(size_t)(r0 + 16) * 4096 + c0 + 16] = acc11[v];
  }
}

// =============================== Launch ====================================

extern "C" void kernel_launch(void* const* d_in, const int* in_sizes, int n_in,
                              void* d_out, int out_size, void* d_ws,
                              size_t ws_size, hipStream_t stream) {
  const float* x = (const float*)d_in[0];    // (8192,1,20,20)
  const float* sig = (const float*)d_in[1];  // (16,2,24,24)
  float* out = (float*)d_out;                // (8192,16,16,16)
  float* w = (float*)d_ws;

  // workspace layout (floats)
  float* Gre = w + 0;
  float* Gim = w + 18432;
  float* WTre = w + 36864;
  float* WTim = w + 55296;
  float* Sre = w + 73728;
  float* Sim = w + 92160;
  const size_t BIG = 110592;
  const size_t PL = (size_t)NCH * PLANE;  // 5,308,416 floats per plane
  float* Bre = w + BIG;
  float* Bim = Bre + PL;
  float* Tre = Bim + PL;
  float* Tim = Tre + PL;
  float* Ure = Tim + PL;
  float* Uim = Ure + PL;
  // Stage-D buffers alias the B region (only used after Horner finishes)
  float* Pre = w + BIG;
  float* Pim = Pre + MAPN;
  float* Qre = Pim + MAPN;
  float* Qim = Qre + MAPN;
  float* Wm = Qim + MAPN;  // 400x4096

  // Stage A
  k_fft_filter<<<32, 576, 0, stream>>>(sig, Gre, Gim);
  k_ifft_window<<<32, 576, 0, stream>>>(Gre, Gim, WTre, WTim);
  k_fft_shift<<<32, 576, 0, stream>>>(WTre, WTim, Sre, Sim);

  // Stage B
  const int nElem = NCH * PLANE;
  k_build_B<<<nElem / 256, 256, 0, stream>>>(Sre, Sim, Bre, Bim);
  k_init_T<<<nElem / 256, 256, 0, stream>>>(Bre, Bim, Tre, Tim);

  // Stage C: expm via Taylor-Horner (N=12) + 10 squarings
  float *tr = Tre, *ti = Tim, *ur = Ure, *ui = Uim;
  for (int k = 11; k >= 1; --k) {
    cgemm_wmma<<<NCH * 81, 256, 0, stream>>>(Bre, Bim, tr, ti, ur, ui,
                                             1.0f / (float)k, 1);
    float* tmp;
    tmp = tr; tr = ur; ur = tmp;
    tmp = ti; ti = ui; ui = tmp;
  }
  for (int s = 0; s < EXPM_S; ++s) {
    cgemm_wmma<<<NCH * 81, 256, 0, stream>>>(tr, ti, tr, ti, ur, ui, 1.0f, 0);
    float* tmp;
    tmp = tr; tr = ur; ur = tmp;
    tmp = ti; ti = ui; ui = tmp;
  }

  // Stage D: slice/roll + ifft2(-2,-1) + fft2(-4,-3), pack real into W
  const int mblk = MAPN / 256;  // 6400
  k_slice_roll<<<mblk, 256, 0, stream>>>(tr, ti, Pre, Pim);
  // ifft over i1 (N=20, inner=1)
  k_dft<<<mblk, 256, 0, stream>>>(Pre, Pim, Qre, Qim, 20, 1, 1.0f, 0.05f);
  // ifft over i0 (N=20, inner=20)
  k_dft<<<mblk, 256, 0, stream>>>(Qre, Qim, Pre, Pim, 20, 20, 1.0f, 0.05f);
  // fft over o1 (N=16, inner=400)
  k_dft<<<mblk, 256, 0, stream>>>(Pre, Pim, Qre, Qim, 16, 400, -1.0f, 1.0f);
  // fft over o0 (N=16, inner=6400)
  k_dft<<<mblk, 256, 0, stream>>>(Qre, Qim, Pre, Pim, 16, 6400, -1.0f, 1.0f);
  k_pack_W<<<mblk, 256, 0, stream>>>(Pre, Wm);

  // Stage E: Y = X @ W
  dim3 grid(4096 / 64, 8192 / 128);
  sgemm_wmma_async<<<grid, 256, 0, stream>>>(x, Wm, out);
}